// ChirpTextureSynth_5050881540237
// MI455X (gfx1250) — compile-verified
//
#include <hip/hip_runtime.h>
#include <math.h>

// ---- problem constants ----
#define SR_F        44100.0
#define N_SAMPLES   131072      // 2^17
#define N_GRAINS    512
#define GRAIN_N     4096
#define BATCH       32
#define LN2         0.6931471805599453
#define L2E_F       1.44269504088896340736f

typedef __attribute__((ext_vector_type(8))) float v8f;
typedef __attribute__((ext_vector_type(2))) float v2f;

#if __has_builtin(__builtin_amdgcn_wmma_f32_16x16x4_f32)
#define HAVE_WMMA4 1
#else
#define HAVE_WMMA4 0
#endif

// sin(2*pi*x): V_SIN_F32 takes input in revolutions on this ISA.
__device__ __forceinline__ float sin2pi(float x) {
#if __has_builtin(__builtin_amdgcn_sinf)
  return __builtin_amdgcn_sinf(x);
#else
  return __sinf(x * 6.28318530717958647692f);
#endif
}

__device__ __forceinline__ float fract_(float x) {
#if __has_builtin(__builtin_amdgcn_fractf)
  return __builtin_amdgcn_fractf(x);
#else
  return x - floorf(x);
#endif
}

__device__ __forceinline__ float exp2_(float x) {
#if __has_builtin(__builtin_amdgcn_exp2f)
  return __builtin_amdgcn_exp2f(x);
#else
  return exp2f(x);
#endif
}

// ---------------------------------------------------------------------------
// Setup: per-(batch,grain) closed-form constants, double precision.
//   inst_freq[t] = (f0/SR)*2^(b*(t-2048)),  b = gamma/SR
//   P[t] = cumsum = C1 * expm1(a*(t+1)),    a = b*ln2
//   C1   = (f0/SR)*2^(-2048 b)/expm1(a)
// Also: amp[g] normalized by max_g, folded with 1/sqrt(f0) into scale[b,g].
// ---------------------------------------------------------------------------
__global__ __launch_bounds__(N_GRAINS) void setup_kernel(
    const float* __restrict__ theta_d, const float* __restrict__ theta_s,
    const float* __restrict__ u, float* __restrict__ a_batch,
    float* __restrict__ c1s, float* __restrict__ scales) {
  const int b = blockIdx.x;
  const int g = threadIdx.x;

  const double d  = (double)theta_d[b];
  const double sl = (double)theta_s[b] * 2.0 - 1.0;          // theta_slope
  const double typical_slope = SR_F / (12.0 * 256.0);        // SR/(Q*HOP)
  const double gamma = tan(0.95 * sl * M_PI * 0.5) * typical_slope * 0.25;
  double bb = gamma / SR_F;          // log2 cycles growth per sample
  double a  = bb * LN2;              // natural-log growth per sample
  if (fabs(a) < 1e-12) {             // keep C1 finite; poly path is exact here
    a  = (a >= 0.0) ? 1e-12 : -1e-12;
    bb = a / LN2;
  }

  // amplitude envelope over grains, normalized by its max
  const double offset = 0.25 * d + 0.75 * d * d;
  const double sig_op = (1.0 - d) * (double)N_GRAINS *
                        ((double)g / (double)N_GRAINS - offset);
  const double amp = 1.0 - 1.0 / (1.0 + exp(-2.0 * sig_op));

  __shared__ float smax[N_GRAINS];
  smax[g] = (float)amp;
  __syncthreads();
  for (int st = N_GRAINS / 2; st > 0; st >>= 1) {
    if (g < st) smax[g] = fmaxf(smax[g], smax[g + st]);
    __syncthreads();
  }
  const double ampmax = (double)smax[0];

  const double uu  = (double)u[b * N_GRAINS + g];
  const double f0  = exp2(uu * 5.0 + 8.0);                   // [256, 8192] Hz
  const double pre = (f0 / SR_F) * exp2(-2048.0 * bb);
  const double C1  = pre / expm1(a);

  const int gi = b * N_GRAINS + g;
  c1s[gi]    = (float)C1;
  scales[gi] = (float)((amp / ampmax) / sqrt(f0));
  if (g == 0) a_batch[b] = (float)a;
}

// ---------------------------------------------------------------------------
// Synthesis: one block = 1024 contiguous samples of one grain.
// Phase via closed form (no cumsum), sin via hardware V_SIN_F32 on fractional
// cycles, scatter-add with fire-and-forget global_atomic_add_f32 (coalesced,
// no wrap needed: onset + t < N_SAMPLES by construction).
// ---------------------------------------------------------------------------
__global__ __launch_bounds__(256) void synth_kernel(
    const int* __restrict__ onsets, const float* __restrict__ a_batch,
    const float* __restrict__ c1s, const float* __restrict__ scales,
    float* __restrict__ x) {
  const int b  = blockIdx.z;
  const int g  = blockIdx.y;
  const int gi = b * N_GRAINS + g;
  const float a  = a_batch[b];     // block-uniform -> scalar loads
  const float C1 = c1s[gi];
  const float sc = scales[gi];
  const int onset = onsets[gi];
  float* __restrict__ xb = x + (size_t)b * N_SAMPLES + onset;
  const int base = blockIdx.x * 1024 + threadIdx.x;

#pragma unroll
  for (int k = 0; k < 4; ++k) {
    const int t = base + k * 256;
    const float uu = a * (float)(t + 1);
    // stable expm1: exp2 path for large |u|, 5th-order poly for small |u|
    const float e2 = exp2_(uu * L2E_F) - 1.0f;
    float p = fmaf(uu, 1.0f / 120.0f, 1.0f / 24.0f);
    p = fmaf(uu, p, 1.0f / 6.0f);
    p = fmaf(uu, p, 0.5f);
    p = fmaf(uu, p, 1.0f);
    p *= uu;
    const float em = (fabsf(uu) < 0.25f) ? p : e2;
    const float P  = C1 * em;                        // phase in *cycles*
    const float sph = sin2pi(fract_(P));             // sin(2*pi*P)
    const float sw  = sin2pi((float)t * (1.0f / 8192.0f)); // sin(pi*t/4096)
    unsafeAtomicAdd(&xb[t], sph * sw * sw * sc);
  }
}

// ---------------------------------------------------------------------------
// Sum of squares per batch. WMMA f32 16x16x4 with A = ones: every row of D
// accumulates the column sums of B, so C row 0 reduced across lanes 0..15
// yields the total of all 64 squared values per issue.
// ---------------------------------------------------------------------------
__global__ __launch_bounds__(256) void sumsq_kernel(
    const float* __restrict__ x, float* __restrict__ norms) {
  const int b = blockIdx.y;
  const float2* __restrict__ xb2 =
      (const float2*)(x + (size_t)b * N_SAMPLES);
  const int lane  = threadIdx.x & 31;
  const int wave  = threadIdx.x >> 5;
  const int base2 = blockIdx.x * 2048;   // 4096 floats per block
  float wsum;

#if HAVE_WMMA4
  v8f acc = {};
  const v2f ones = {1.0f, 1.0f};
  for (int it = 0; it < 8; ++it) {
    float2 v = xb2[base2 + it * 256 + wave * 32 + lane];
    v2f bm;
    bm.x = v.x * v.x;
    bm.y = v.y * v.y;
    acc = __builtin_amdgcn_wmma_f32_16x16x4_f32(
        false, ones, false, bm, (short)0, acc, false, false);
  }
  // row 0 of C lives in acc[0], lanes 0..15 (row 8 duplicate in lanes 16..31)
  float r = acc[0];
  r += __shfl_xor(r, 1, 32);
  r += __shfl_xor(r, 2, 32);
  r += __shfl_xor(r, 4, 32);
  r += __shfl_xor(r, 8, 32);
  wsum = r;
#else
  float r = 0.0f;
  for (int it = 0; it < 8; ++it) {
    float2 v = xb2[base2 + it * 256 + wave * 32 + lane];
    r = fmaf(v.x, v.x, r);
    r = fmaf(v.y, v.y, r);
  }
  for (int off = 16; off >= 1; off >>= 1) r += __shfl_xor(r, off, 32);
  wsum = r;
#endif

  __shared__ float ws[8];
  if (lane == 0) ws[wave] = wsum;
  __syncthreads();
  if (threadIdx.x == 0) {
    float s = 0.0f;
    for (int w = 0; w < 8; ++w) s += ws[w];
    unsafeAtomicAdd(&norms[b], s);
  }
}

__global__ __launch_bounds__(256) void normalize_kernel(
    float* __restrict__ x, const float* __restrict__ norms) {
  const int b = blockIdx.y;
  const float inv = rsqrtf(norms[b]);     // x / ||x|| = x * rsqrt(sum x^2)
  float4* p = (float4*)(x + (size_t)b * N_SAMPLES) +
              blockIdx.x * 256 + threadIdx.x;
  float4 v = *p;
  v.x *= inv; v.y *= inv; v.z *= inv; v.w *= inv;
  *p = v;
}

extern "C" void kernel_launch(void* const* d_in, const int* in_sizes, int n_in,
                              void* d_out, int out_size, void* d_ws,
                              size_t ws_size, hipStream_t stream) {
  (void)in_sizes; (void)n_in; (void)ws_size;
  const float* theta_d = (const float*)d_in[0];
  const float* theta_s = (const float*)d_in[1];
  const float* u       = (const float*)d_in[2];
  const int*   onsets  = (const int*)d_in[3];
  float* x = (float*)d_out;

  float* ws      = (float*)d_ws;
  float* norms   = ws;                            // 32
  float* a_batch = ws + 32;                       // 32
  float* c1s     = ws + 64;                       // 32*512
  float* scales  = ws + 64 + BATCH * N_GRAINS;    // 32*512

  hipMemsetAsync(d_out, 0, (size_t)out_size * sizeof(float), stream);
  hipMemsetAsync(norms, 0, BATCH * sizeof(float), stream);

  setup_kernel<<<BATCH, N_GRAINS, 0, stream>>>(theta_d, theta_s, u,
                                               a_batch, c1s, scales);

  dim3 sg(GRAIN_N / 1024, N_GRAINS, BATCH);       // (4, 512, 32)
  synth_kernel<<<sg, 256, 0, stream>>>(onsets, a_batch, c1s, scales, x);

  sumsq_kernel<<<dim3(N_SAMPLES / 4096, BATCH), 256, 0, stream>>>(x, norms);
  normalize_kernel<<<dim3(N_SAMPLES / 1024, BATCH), 256, 0, stream>>>(x, norms);
}